// CLEVRThreeDEmbeddingWithSinCosNumbers_90452011253991
// MI455X (gfx1250) — compile-verified
//
#include <hip/hip_runtime.h>
#include <hip/hip_bf16.h>

// ---------------------------------------------------------------------------
// CLEVR mixed-source embedding:
//   text/number/3d tokens -> pure gathers (memory bound, ~512MB -> ~22us HBM)
//   image tokens          -> codebook gather + [n_img,256]x[256,2048] f32 GEMM
// Image tokens are compacted into a workspace list so the WMMA GEMM runs only
// on the ~33% of tokens that need it (11.5 GFLOP instead of 34 GFLOP dense).
// GEMM uses V_WMMA_F32_16X16X4_F32 (exact f32 precision, matches reference),
// 4 concurrent N-tile accumulator chains per wave for XDL pipelining and 4x
// A-fragment reuse per LDS load.
// ---------------------------------------------------------------------------

typedef __attribute__((ext_vector_type(2))) float v2f;
typedef __attribute__((ext_vector_type(8))) float v8f;

#define ATO        32000
#define NUM_END    33000   // ATO + SCO
#define TD_END     33100   // VQ_START / IMG_OFFSET
#define VQ_END     49484
#define EMBED_DIM  2048
#define VQ_DIM     256
#define LDSA_STRIDE 260    // 256 + 4 pad: LDS bank = (4*t) % 64, conflict-free

// ---------------------------------------------------------------------------
// Kernel 1: one wave (32 lanes) per token. Gather rows for the three table
// sources with float4 (global_load_b128/global_store_b128); append image
// tokens to the compacted work list.
// ---------------------------------------------------------------------------
__global__ void clevr_gather_kernel(const int* __restrict__ x,
                                    const float* __restrict__ token_emb,
                                    const float* __restrict__ added_emb,
                                    const float* __restrict__ numbers_emb,
                                    float* __restrict__ out,
                                    int* __restrict__ counter,
                                    int* __restrict__ list_pos,
                                    int* __restrict__ list_row,
                                    int n_tokens) {
  const int wave = threadIdx.x >> 5;
  const int lane = threadIdx.x & 31;
  const int t = blockIdx.x * 8 + wave;
  if (t >= n_tokens) return;

  const int tok = x[t];
  const float* src = nullptr;
  if (tok >= 0 && tok < ATO) {
    src = token_emb + (size_t)tok * EMBED_DIM;
  } else if (tok >= ATO && tok < NUM_END) {
    src = numbers_emb + (size_t)(tok - ATO) * EMBED_DIM;
  } else if (tok >= NUM_END && tok < TD_END) {
    src = added_emb + (size_t)(tok - NUM_END) * EMBED_DIM;
  } else if (tok >= TD_END && tok < VQ_END) {
    if (lane == 0) {
      int slot = atomicAdd(counter, 1);
      list_pos[slot] = t;
      list_row[slot] = tok - TD_END;
    }
    return;  // row written by the WMMA projection kernel
  }

  float4* dst = (float4*)(out + (size_t)t * EMBED_DIM);
  if (src) {
    const float4* s4 = (const float4*)src;
#pragma unroll
    for (int j = 0; j < 16; ++j) dst[lane + 32 * j] = s4[lane + 32 * j];
  } else {
    const float4 z = {0.f, 0.f, 0.f, 0.f};
#pragma unroll
    for (int j = 0; j < 16; ++j) dst[lane + 32 * j] = z;
  }
}

// ---------------------------------------------------------------------------
// Kernel 2: per block, one 16-token M-tile of compacted image tokens.
// A (16x256 codebook rows) staged in LDS; B streamed from proj_w (L2-resident,
// 2MB). 8 waves x 256 output columns; each wave runs 4 batches of 4 N-tiles,
// each batch = 64 K-steps with 4 independent V_WMMA_F32_16X16X4_F32 chains.
// Grid sized for the worst case; blocks beyond ceil(count/16) exit early
// (fixed grid keeps graph capture happy).
//
// ISA operand layouts (wave32):
//   A 16x4 f32 : lane l, vgpr j -> A[l%16][2*(l/16)+j]
//   B  4x16 f32: lane l, vgpr j -> B[2*(l/16)+j][l%16]   (B[k][n]=proj_w[n][k])
//   C/D 16x16  : lane l, vgpr i -> D[i + 8*(l/16)][l%16]
// ---------------------------------------------------------------------------
__global__ void clevr_img_proj_kernel(const float* __restrict__ codebook,
                                      const float* __restrict__ proj_w,
                                      const int* __restrict__ counter,
                                      const int* __restrict__ list_pos,
                                      const int* __restrict__ list_row,
                                      float* __restrict__ out) {
  __shared__ float As[16 * LDSA_STRIDE];
  __shared__ int s_pos[16];
  __shared__ int s_row[16];

  const int count = *counter;
  const int tiles = (count + 15) >> 4;
  if ((int)blockIdx.x >= tiles) return;
  const int tile_base = blockIdx.x * 16;
  const bool tile_full = (tile_base + 16 <= count);  // uniform: all rows valid

  if (threadIdx.x < 16) {
    const int g = tile_base + threadIdx.x;
    if (g < count) {
      s_pos[threadIdx.x] = list_pos[g];
      s_row[threadIdx.x] = list_row[g];
    } else {
      s_pos[threadIdx.x] = -1;
      s_row[threadIdx.x] = -1;  // tail padding row -> zeros, no store
    }
  }
  __syncthreads();

  // Stage A tile: thread i handles token i/16, 16-float chunk i%16.
  {
    const int tt = threadIdx.x >> 4;
    const int cc = threadIdx.x & 15;
    const int row = s_row[tt];
    float4* d4 = (float4*)(&As[tt * LDSA_STRIDE + cc * 16]);
    if (row >= 0) {
      const float4* s4 = (const float4*)(codebook + (size_t)row * VQ_DIM + cc * 16);
      d4[0] = s4[0]; d4[1] = s4[1]; d4[2] = s4[2]; d4[3] = s4[3];
    } else {
      const float4 z = {0.f, 0.f, 0.f, 0.f};
      d4[0] = z; d4[1] = z; d4[2] = z; d4[3] = z;
    }
  }
  __syncthreads();

  const int wave = threadIdx.x >> 5;
  const int lane = threadIdx.x & 31;
  const int hl   = lane >> 4;    // lane half: selects K pair
  const int l16  = lane & 15;    // A row (M) / B column (N) within tile
  const int koff = hl * 2;

  const float* aBase = &As[l16 * LDSA_STRIDE + koff];

  // 16 N-tiles per wave, processed 4 at a time (4 independent WMMA chains).
  for (int ntb = 0; ntb < 4; ++ntb) {
    const int n_base = wave * 256 + ntb * 64;
    const float* bBase = proj_w + (size_t)(n_base + l16) * VQ_DIM + koff;

    v8f c0 = {0.f, 0.f, 0.f, 0.f, 0.f, 0.f, 0.f, 0.f};
    v8f c1 = c0, c2 = c0, c3 = c0;

#pragma unroll 4
    for (int k = 0; k < VQ_DIM; k += 4) {
      v2f a  = *(const v2f*)(aBase + k);                 // ds_load_b64, reused x4
      v2f b0 = *(const v2f*)(bBase + k);                 // global_load_b64, L2 hit
      v2f b1 = *(const v2f*)(bBase + k + 16 * VQ_DIM);   //  (+16KB imm offsets)
      v2f b2 = *(const v2f*)(bBase + k + 32 * VQ_DIM);
      v2f b3 = *(const v2f*)(bBase + k + 48 * VQ_DIM);
      c0 = __builtin_amdgcn_wmma_f32_16x16x4_f32(false, a, false, b0, (short)0, c0, false, false);
      c1 = __builtin_amdgcn_wmma_f32_16x16x4_f32(false, a, false, b1, (short)0, c1, false, false);
      c2 = __builtin_amdgcn_wmma_f32_16x16x4_f32(false, a, false, b2, (short)0, c2, false, false);
      c3 = __builtin_amdgcn_wmma_f32_16x16x4_f32(false, a, false, b3, (short)0, c3, false, false);
    }

    if (tile_full) {
      // Common path: no divergence, plain global_store_b32 scatter.
#pragma unroll
      for (int i = 0; i < 8; ++i) {
        const size_t rowbase = (size_t)s_pos[i + hl * 8] * EMBED_DIM + l16;
        out[rowbase + n_base +  0] = c0[i];
        out[rowbase + n_base + 16] = c1[i];
        out[rowbase + n_base + 32] = c2[i];
        out[rowbase + n_base + 48] = c3[i];
      }
    } else {
      // At most one block takes this path (tail tile).
#pragma unroll
      for (int i = 0; i < 8; ++i) {
        const int pos = s_pos[i + hl * 8];
        if (pos >= 0) {
          const size_t rowbase = (size_t)pos * EMBED_DIM + l16;
          out[rowbase + n_base +  0] = c0[i];
          out[rowbase + n_base + 16] = c1[i];
          out[rowbase + n_base + 32] = c2[i];
          out[rowbase + n_base + 48] = c3[i];
        }
      }
    }
  }
}

extern "C" void kernel_launch(void* const* d_in, const int* in_sizes, int n_in,
                              void* d_out, int out_size, void* d_ws, size_t ws_size,
                              hipStream_t stream) {
  const int*   x           = (const int*)d_in[0];
  const float* token_emb   = (const float*)d_in[1];
  const float* added_emb   = (const float*)d_in[2];
  const float* numbers_emb = (const float*)d_in[3];
  const float* codebook    = (const float*)d_in[4];
  const float* proj_w      = (const float*)d_in[5];
  float* out = (float*)d_out;

  const int n_tokens = in_sizes[0];  // 4 * 8192 = 32768

  // Workspace: [0..15] counter (+pad), then compacted pos list, then row list.
  int* counter  = (int*)d_ws;
  int* list_pos = counter + 16;
  int* list_row = list_pos + n_tokens;

  hipMemsetAsync(counter, 0, 16 * sizeof(int), stream);

  clevr_gather_kernel<<<(n_tokens + 7) / 8, 256, 0, stream>>>(
      x, token_emb, added_emb, numbers_emb, out,
      counter, list_pos, list_row, n_tokens);

  const int max_tiles = (n_tokens + 15) / 16;  // worst case: all image tokens
  clevr_img_proj_kernel<<<max_tiles, 256, 0, stream>>>(
      codebook, proj_w, counter, list_pos, list_row, out);
}